// GraphNudger_72524817760795
// MI455X (gfx1250) — compile-verified
//
#include <hip/hip_runtime.h>

// Problem constants (fixed by the reference file)
#define S_DIM  10000
#define D_DIM  100
#define DPAD   112          // 7 WMMA N-tiles of 16
#define NTILES 7
#define FDIM   2048
#define ETA_F  0.01f
#define KC     100          // K-chunk staged in LDS (divides S_DIM/NSPLIT, %4==0)
#define NSPLIT 10           // split-K factor for occupancy
#define CH4    (KC * DPAD / 4)          // float4s per chunk = 2800
#define RITER  ((CH4 + 255) / 256)      // async b128 ops per thread per chunk = 11

typedef float v2f __attribute__((ext_vector_type(2)));
typedef float v4f __attribute__((ext_vector_type(4)));
typedef float v8f __attribute__((ext_vector_type(8)));
typedef int   v4i __attribute__((ext_vector_type(4)));

// Pointer-to-AS-qualified-int4 types for the async-to-LDS builtin
typedef __attribute__((address_space(1))) v4i* gv4i_p;   // global (device) int4*
typedef __attribute__((address_space(3))) v4i* lv4i_p;   // LDS int4*

#if defined(__gfx1250__) && \
    __has_builtin(__builtin_amdgcn_global_load_async_to_lds_b128) && \
    __has_builtin(__builtin_amdgcn_s_wait_asynccnt)
#define USE_ASYNC_LDS 1
#else
#define USE_ASYNC_LDS 0
#endif

// K-paired layout: values for (s, d) and (s+1, d) are adjacent -> one b64 LDS load
__device__ __forceinline__ size_t apad_idx(int s, int d) {
    return ((size_t)(s >> 1) * DPAD + d) * 2 + (s & 1);
}

__global__ void zero_kernel(float* apad, size_t n_apad, float* out, size_t n_out) {
    size_t i = (size_t)blockIdx.x * blockDim.x + threadIdx.x;
    if (i < n_apad) apad[i] = 0.0f;
    if (i < n_out)  out[i]  = 0.0f;
}

__global__ void scatter_kernel(const float* __restrict__ w,
                               const int* __restrict__ src,
                               const int* __restrict__ dst,
                               float* __restrict__ apad, int E) {
    int e = blockIdx.x * blockDim.x + threadIdx.x;
    if (e < E) {
        int d = src[e];              // disease node
        int s = dst[e];              // sign node
        atomicAdd(&apad[apad_idx(s, d)], w[e]);
    }
}

__global__ void norm_kernel(const float* __restrict__ grad, float* __restrict__ g) {
    __shared__ float partial[8];
    const int b   = blockIdx.x;
    const int tid = threadIdx.x;
    const v4f* row = (const v4f*)(grad + (size_t)b * FDIM);
    float sum = 0.0f;
    for (int i = tid; i < FDIM / 4; i += 256) {
        v4f v = __builtin_nontemporal_load(row + i);   // streamed once: keep out of L2
        sum += v[0] * v[0] + v[1] * v[1] + v[2] * v[2] + v[3] * v[3];
    }
    for (int off = 16; off > 0; off >>= 1) sum += __shfl_xor(sum, off, 32);
    if ((tid & 31) == 0) partial[tid >> 5] = sum;
    __syncthreads();
    if (tid == 0) {
        float t = 0.0f;
        for (int j = 0; j < 8; ++j) t += partial[j];
        g[b] = sqrtf(t);
    }
}

// Stage one KC x DPAD chunk of Apad into LDS.
// Async path: every wave issues exactly RITER async-b128 ops (uniform ASYNCcnt).
__device__ __forceinline__ void stage_chunk(float* dst_lds, const float* src, int tid) {
#if USE_ASYNC_LDS
    const v4i* s4 = (const v4i*)src;
    v4i*       d4 = (v4i*)dst_lds;
    #pragma unroll
    for (int r = 0; r < RITER; ++r) {
        int i = tid + r * 256;
        if (i < CH4) {
            __builtin_amdgcn_global_load_async_to_lds_b128(
                (gv4i_p)(s4 + i), (lv4i_p)(d4 + i),
                /*offset=*/0, /*cpol=*/0);
        }
    }
#else
    const float4* s4 = (const float4*)src;
    float4*       d4 = (float4*)dst_lds;
    for (int i = tid; i < CH4; i += 256) d4[i] = s4[i];
#endif
}

// 8 waves/block, each wave owns one 16-row M-tile and all 7 N-tiles.
// grid = (B/128, NSPLIT); split-K partials accumulated with f32 atomics.
// Double-buffered LDS staging of the L2-resident A matrix.
__global__ void __launch_bounds__(256, 1)
gemm_kernel(const float* __restrict__ sims, const float* __restrict__ apad,
            const float* __restrict__ g, float* __restrict__ out) {
    __shared__ float ldsA[2][KC * DPAD];   // 2 x 44800 B, paired-K layout

    const int tid  = threadIdx.x;
    const int lane = tid & 31;
    const int wave = tid >> 5;
    const int l15  = lane & 15;
    const int hal  = lane >> 4;         // 0 or 1
    const int koff = hal * 2;           // A frag: lanes<16 hold K=0,1; lanes>=16 hold K=2,3

    const int b0      = (blockIdx.x * 8 + wave) * 16;
    const int s_begin = blockIdx.y * (S_DIM / NSPLIT);
    const int NCH     = (S_DIM / NSPLIT) / KC;   // 10 chunks per split

    // A-operand source: row b0+l15 of sims, columns (s + koff, s + koff + 1)
    const float* simsrow = sims + (size_t)(b0 + l15) * S_DIM + koff;

    v8f acc[NTILES];
    for (int t = 0; t < NTILES; ++t)
        acc[t] = (v8f){0.f, 0.f, 0.f, 0.f, 0.f, 0.f, 0.f, 0.f};

    // Prefetch chunk 0
    stage_chunk(ldsA[0], apad + (size_t)s_begin * DPAD, tid);

    for (int c = 0; c < NCH; ++c) {
        const int cur = c & 1;

        // Prefetch next chunk into the other buffer (its last readers finished
        // at the end of iteration c-1, which closed with a barrier).
        if (c + 1 < NCH)
            stage_chunk(ldsA[cur ^ 1],
                        apad + (size_t)(s_begin + (c + 1) * KC) * DPAD, tid);

#if USE_ASYNC_LDS
        // Release chunk c while chunk c+1 (RITER ops/wave) stays in flight.
        if (c + 1 < NCH) __builtin_amdgcn_s_wait_asynccnt(RITER);
        else             __builtin_amdgcn_s_wait_asynccnt(0);
#endif
        __syncthreads();

        const float* ldsbuf = ldsA[cur];
        const int s0 = s_begin + c * KC;
        for (int kk = 0; kk < KC; kk += 4) {
            // A fragment (16x4 f32 layout): one NT 8-byte global load per lane
            v2f a = __builtin_nontemporal_load((const v2f*)(simsrow + s0 + kk));

            // B fragments: rows (kk+koff, kk+koff+1) x cols nt*16+l15, one b64 each
            const float2* brow = (const float2*)ldsbuf + ((kk + koff) >> 1) * DPAD + l15;
            #pragma unroll
            for (int nt = 0; nt < NTILES; ++nt) {
                const float2 bv = brow[nt * 16];
                v2f b; b[0] = bv.x; b[1] = bv.y;
                acc[nt] = __builtin_amdgcn_wmma_f32_16x16x4_f32(
                    /*neg_a=*/false, a, /*neg_b=*/false, b,
                    /*c_mod=*/(short)0, acc[nt], /*reuse_a=*/false, /*reuse_b=*/false);
            }
        }
        __syncthreads();   // all waves done reading buffer `cur` before it is restaged
    }

    // Epilogue: C/D layout VGPR j -> (M = j + 8*hal, N = l15); scale by ETA*g, atomic split-K add
    const int mbase = hal * 8;
    float scale[8];
    #pragma unroll
    for (int j = 0; j < 8; ++j) scale[j] = ETA_F * g[b0 + mbase + j];
    #pragma unroll
    for (int nt = 0; nt < NTILES; ++nt) {
        const int col = nt * 16 + l15;
        if (col < D_DIM) {
            #pragma unroll
            for (int j = 0; j < 8; ++j) {
                atomicAdd(&out[(size_t)(b0 + mbase + j) * D_DIM + col],
                          scale[j] * acc[nt][j]);
            }
        }
    }
}

extern "C" void kernel_launch(void* const* d_in, const int* in_sizes, int n_in,
                              void* d_out, int out_size, void* d_ws, size_t ws_size,
                              hipStream_t stream) {
    const float* grad = (const float*)d_in[0];   // [B, FDIM]
    const float* sims = (const float*)d_in[1];   // [B, S]
    const float* ew   = (const float*)d_in[2];   // [E]
    const int*   esrc = (const int*)d_in[3];     // [E] disease ids
    const int*   edst = (const int*)d_in[4];     // [E] sign ids

    const int E = in_sizes[2];
    const int B = in_sizes[0] / FDIM;            // 8192

    float* apad = (float*)d_ws;                               // S_DIM*DPAD floats (4.48 MB)
    float* g    = apad + (size_t)S_DIM * DPAD;                // B floats
    float* out  = (float*)d_out;                              // [B, D_DIM]

    const size_t n_apad = (size_t)S_DIM * DPAD;
    const size_t n_out  = (size_t)B * D_DIM;
    const size_t n_max  = n_apad > n_out ? n_apad : n_out;

    zero_kernel<<<(unsigned)((n_max + 255) / 256), 256, 0, stream>>>(apad, n_apad, out, n_out);
    scatter_kernel<<<(E + 255) / 256, 256, 0, stream>>>(ew, esrc, edst, apad, E);
    norm_kernel<<<B, 256, 0, stream>>>(grad, g);

    dim3 grid(B / 128, NSPLIT);                  // 8 waves * 16 rows = 128 rows per block
    gemm_kernel<<<grid, 256, 0, stream>>>(sims, apad, g, out);
}